// STGCNLayer_15848429322683
// MI455X (gfx1250) — compile-verified
//
#include <hip/hip_runtime.h>
#include <hip/hip_bf16.h>

// STGCN layer for MI455X (gfx1250, wave32, WMMA f32 16x16x4)
// T=16, N=20000, E=320000, C=128

#define TT 16
#define NN 20000
#define EE 320000
#define CC 128
#define KK 384            // 3*C (fused temporal taps)
#define PANEL_STRIDE 388  // 384 + 4 pad (floats), keeps rows 8B-aligned
#define BN_EPS 1e-5f

typedef __attribute__((ext_vector_type(2))) float v2f;
typedef __attribute__((ext_vector_type(8))) float v8f;

static inline int cdiv(int a, int b) { return (a + b - 1) / b; }

// Hardware f32 atomic add (global_atomic_add_f32), no CAS loop.
__device__ __forceinline__ void atomicAddF32(float* p, float v) {
    __hip_atomic_fetch_add(p, v, __ATOMIC_RELAXED, __HIP_MEMORY_SCOPE_AGENT);
}

// ---------------- precompute ----------------

__global__ __launch_bounds__(256) void k_deg_init(float* __restrict__ deg) {
    int n = blockIdx.x * 256 + threadIdx.x;
    if (n < NN) deg[n] = 1.0f;  // self-loop weight
}

__global__ __launch_bounds__(256) void k_deg_accum(const int* __restrict__ dst,
                                                   const float* __restrict__ w,
                                                   float* __restrict__ deg) {
    int e = blockIdx.x * 256 + threadIdx.x;
    if (e < EE) atomicAddF32(&deg[dst[e]], w[e]);
}

__global__ __launch_bounds__(256) void k_dinv(const float* __restrict__ deg,
                                              float* __restrict__ dinv,
                                              float* __restrict__ selfnorm) {
    int n = blockIdx.x * 256 + threadIdx.x;
    if (n >= NN) return;
    float d = deg[n];
    float di = (d > 0.0f) ? rsqrtf(fmaxf(d, 1e-12f)) : 0.0f;
    dinv[n] = di;
    selfnorm[n] = di * di;  // self-loop norm (w = 1)
}

__global__ __launch_bounds__(256) void k_norm(const int* __restrict__ src,
                                              const int* __restrict__ dst,
                                              const float* __restrict__ w,
                                              const float* __restrict__ dinv,
                                              float* __restrict__ norm) {
    int e = blockIdx.x * 256 + threadIdx.x;
    if (e < EE) norm[e] = dinv[src[e]] * w[e] * dinv[dst[e]];
}

// Wcat[c, s*C + i] = sum_j Wg[c,j] * Wt[j,i,s]   (Wt is [C][C][3])
__global__ __launch_bounds__(256) void k_wcat(const float* __restrict__ Wg,
                                              const float* __restrict__ Wt,
                                              float* __restrict__ Wcat) {
    int idx = blockIdx.x * 256 + threadIdx.x;
    if (idx >= CC * KK) return;
    int c = idx / KK;
    int k = idx % KK;
    int s = k / CC;   // temporal tap 0..2
    int i = k % CC;   // input channel
    float acc = 0.0f;
    const float* wg = Wg + (size_t)c * CC;
    const float* wt = Wt + (size_t)i * 3 + s;
    #pragma unroll 4
    for (int j = 0; j < CC; ++j) acc += wg[j] * wt[(size_t)j * CC * 3];
    Wcat[idx] = acc;
}

__global__ __launch_bounds__(128) void k_b2(const float* __restrict__ Wg,
                                            const float* __restrict__ bt,
                                            float* __restrict__ b2) {
    int c = threadIdx.x;
    float acc = 0.0f;
    for (int j = 0; j < CC; ++j) acc += Wg[(size_t)c * CC + j] * bt[j];
    b2[c] = acc;
}

// ---------------- fused temporal-conv + GCN-linear GEMM ----------------
// ht[n, c] = sum_k Xcat[n, k] * Wcat[c, k] + b2[c]
// Xcat[n, s*C+i] = x[t+s-1, n, i]  (zero-padded at t boundaries)
// One block = 32 nodes x 128 channels; 8 waves; each wave computes TWO
// 16x16 node-tiles sharing one B fragment (halves Wcat traffic, doubles
// wmma per B load). K = 384.
__global__ __launch_bounds__(256) void k_gemm(const float* __restrict__ x,
                                              const float* __restrict__ Wcat,
                                              const float* __restrict__ b2,
                                              float* __restrict__ ht, int t) {
    __shared__ float panel[32 * PANEL_STRIDE];
    const int tid = threadIdx.x;
    const int n0 = blockIdx.x * 32;

    // Cooperative load of 32x384 X panel (3072 float4 / 256 threads = 12 each)
    #pragma unroll
    for (int j = 0; j < 12; ++j) {
        int idx4 = tid + 256 * j;        // 0..3071
        int r = idx4 / 96;               // node row 0..31 (96 float4 per row)
        int q = idx4 % 96;
        int s = q >> 5;                  // tap 0..2
        int col = (q & 31) << 2;         // channel 0..124 step 4
        int ts = t + s - 1;
        float4 v = make_float4(0.f, 0.f, 0.f, 0.f);
        if (ts >= 0 && ts < TT) {
            v = *(const float4*)(x + (((size_t)ts * NN + (n0 + r)) * CC + col));
        }
        float* dp = &panel[r * PANEL_STRIDE + s * CC + col];
        dp[0] = v.x; dp[1] = v.y; dp[2] = v.z; dp[3] = v.w;
    }
    __syncthreads();

    const int lane = tid & 31;
    const int wave = tid >> 5;
    const int c0 = wave * 16;      // channel tile base
    const int lr = lane & 15;      // A: M row / B: N col
    const int lh = lane >> 4;      // lane half selects K pair (ISA 7.12.2)

    v8f acc0 = {};
    v8f acc1 = {};
    const float* arow0 = &panel[lr * PANEL_STRIDE + 2 * lh];          // nodes n0..n0+15
    const float* arow1 = &panel[(16 + lr) * PANEL_STRIDE + 2 * lh];   // nodes n0+16..n0+31
    const float* brow  = Wcat + (size_t)(c0 + lr) * KK + 2 * lh;      // global, L2-hot

    #pragma unroll 8
    for (int k0 = 0; k0 < KK; k0 += 4) {
        v2f b  = *(const v2f*)(brow + k0);   // B[k0+2*lh..+1][lr] = Wcat[c0+lr][...]
        v2f a0 = *(const v2f*)(arow0 + k0);  // A0[lr][k0+2*lh..+1]
        v2f a1 = *(const v2f*)(arow1 + k0);
        acc0 = __builtin_amdgcn_wmma_f32_16x16x4_f32(
            false, a0, false, b, (short)0, acc0, false, false);
        acc1 = __builtin_amdgcn_wmma_f32_16x16x4_f32(
            false, a1, false, b, (short)0, acc1, false, false);
    }

    // C/D layout: VGPR v, lanes 0-15 -> M=v, lanes 16-31 -> M=v+8; N=lane%16
    float bias = b2[c0 + lr];
    #pragma unroll
    for (int v = 0; v < 8; ++v) {
        int node = n0 + v + 8 * lh;
        ht[(size_t)node * CC + c0 + lr] = acc0[v] + bias;
    }
    #pragma unroll
    for (int v = 0; v < 8; ++v) {
        int node = n0 + 16 + v + 8 * lh;
        ht[(size_t)node * CC + c0 + lr] = acc1[v] + bias;
    }
}

// ---------------- scatter / aggregate ----------------

// agg = ht * selfnorm + bg ; also zero BN stat accumulators
__global__ __launch_bounds__(256) void k_agg_init(const float* __restrict__ ht,
                                                  const float* __restrict__ selfnorm,
                                                  const float* __restrict__ bg,
                                                  float* __restrict__ agg,
                                                  float* __restrict__ sum,
                                                  float* __restrict__ sumsq) {
    int i4 = blockIdx.x * 256 + threadIdx.x;
    const int total4 = NN * CC / 4;
    if (i4 < total4) {
        int i = i4 * 4;
        int n = i / CC;
        int c = i % CC;
        float sn = selfnorm[n];
        float4 h = *(const float4*)(ht + i);
        float4 o;
        o.x = h.x * sn + bg[c + 0];
        o.y = h.y * sn + bg[c + 1];
        o.z = h.z * sn + bg[c + 2];
        o.w = h.w * sn + bg[c + 3];
        *(float4*)(agg + i) = o;
    }
    if (blockIdx.x == 0 && threadIdx.x < CC) {
        sum[threadIdx.x] = 0.0f;
        sumsq[threadIdx.x] = 0.0f;
    }
}

// one wave per edge: gather 128 floats from ht[src], scale, atomic-add to agg[dst]
__global__ __launch_bounds__(256) void k_edges(const float* __restrict__ ht,
                                               const int* __restrict__ src,
                                               const int* __restrict__ dst,
                                               const float* __restrict__ norm,
                                               float* __restrict__ agg) {
    int e = blockIdx.x * 8 + (threadIdx.x >> 5);
    if (e >= EE) return;
    int lane = threadIdx.x & 31;
    int s = src[e], d = dst[e];
    float nw = norm[e];
    float4 v = *((const float4*)(ht + (size_t)s * CC) + lane);
    float* dp = agg + (size_t)d * CC + lane * 4;
    atomicAddF32(dp + 0, v.x * nw);
    atomicAddF32(dp + 1, v.y * nw);
    atomicAddF32(dp + 2, v.z * nw);
    atomicAddF32(dp + 3, v.w * nw);
}

// ---------------- batchnorm ----------------

__global__ __launch_bounds__(128) void k_bnstats(const float* __restrict__ agg,
                                                 float* __restrict__ sum,
                                                 float* __restrict__ sumsq) {
    int c = threadIdx.x;
    int r0 = blockIdx.x * 157;
    int r1 = r0 + 157;
    if (r1 > NN) r1 = NN;
    float s = 0.0f, s2 = 0.0f;
    for (int r = r0; r < r1; ++r) {
        float v = agg[(size_t)r * CC + c];
        s += v;
        s2 += v * v;
    }
    atomicAddF32(&sum[c], s);
    atomicAddF32(&sumsq[c], s2);
}

__global__ __launch_bounds__(256) void k_bnapply(const float* __restrict__ agg,
                                                 const float* __restrict__ sum,
                                                 const float* __restrict__ sumsq,
                                                 const float* __restrict__ gamma,
                                                 const float* __restrict__ beta,
                                                 float* __restrict__ out) {
    int i4 = blockIdx.x * 256 + threadIdx.x;
    const int total4 = NN * CC / 4;
    if (i4 >= total4) return;
    int i = i4 * 4;
    int c = i % CC;
    float4 a = *(const float4*)(agg + i);
    const float invN = 1.0f / (float)NN;
    float va[4] = {a.x, a.y, a.z, a.w};
    float o[4];
    #pragma unroll
    for (int j = 0; j < 4; ++j) {
        float mu = sum[c + j] * invN;
        float var = sumsq[c + j] * invN - mu * mu;
        float inv = rsqrtf(var + BN_EPS);
        float y = gamma[c + j] * (va[j] - mu) * inv + beta[c + j];
        o[j] = fmaxf(y, 0.0f);
    }
    float4 ov = make_float4(o[0], o[1], o[2], o[3]);
    *(float4*)(out + i) = ov;
}

// ---------------- launch ----------------

extern "C" void kernel_launch(void* const* d_in, const int* in_sizes, int n_in,
                              void* d_out, int out_size, void* d_ws, size_t ws_size,
                              hipStream_t stream) {
    const float* x     = (const float*)d_in[0];  // [T,N,C]
    const int*   eidx  = (const int*)d_in[1];    // [2,E]
    const float* ew    = (const float*)d_in[2];  // [E]
    const float* Wt    = (const float*)d_in[3];  // [C,C,3]
    const float* bt    = (const float*)d_in[4];  // [C]
    const float* Wg    = (const float*)d_in[5];  // [C,C]
    const float* bg    = (const float*)d_in[6];  // [C]
    const float* gamma = (const float*)d_in[7];  // [C]
    const float* beta  = (const float*)d_in[8];  // [C]
    float* out = (float*)d_out;

    const int* srcI = eidx;
    const int* dstI = eidx + EE;

    // workspace layout (floats, all float4-aligned offsets)
    float* ws = (float*)d_ws;
    float* deg      = ws;                      // N
    float* dinv     = deg + NN;                // N
    float* selfnorm = dinv + NN;               // N
    float* norm     = selfnorm + NN;           // E
    float* Wcat     = norm + EE;               // C*3C = 49152
    float* b2       = Wcat + CC * KK;          // 128
    float* sum      = b2 + CC;                 // 128
    float* sumsq    = sum + CC;                // 128
    float* ht       = sumsq + CC;              // N*C
    float* agg      = ht + (size_t)NN * CC;    // N*C

    // --- precompute ---
    k_deg_init<<<cdiv(NN, 256), 256, 0, stream>>>(deg);
    k_deg_accum<<<cdiv(EE, 256), 256, 0, stream>>>(dstI, ew, deg);
    k_dinv<<<cdiv(NN, 256), 256, 0, stream>>>(deg, dinv, selfnorm);
    k_norm<<<cdiv(EE, 256), 256, 0, stream>>>(srcI, dstI, ew, dinv, norm);
    k_wcat<<<cdiv(CC * KK, 256), 256, 0, stream>>>(Wg, Wt, Wcat);
    k_b2<<<1, 128, 0, stream>>>(Wg, bt, b2);

    const int gemm_blocks = NN / 32;           // 625 (exact)
    const int elem4_blocks = cdiv(NN * CC / 4, 256);
    const int edge_blocks = cdiv(EE, 8);       // wave per edge

    for (int t = 0; t < TT; ++t) {
        k_gemm<<<gemm_blocks, 256, 0, stream>>>(x, Wcat, b2, ht, t);
        k_agg_init<<<elem4_blocks, 256, 0, stream>>>(ht, selfnorm, bg, agg, sum, sumsq);
        k_edges<<<edge_blocks, 256, 0, stream>>>(ht, srcI, dstI, norm, agg);
        k_bnstats<<<128, 128, 0, stream>>>(agg, sum, sumsq);
        k_bnapply<<<elem4_blocks, 256, 0, stream>>>(agg, sum, sumsq, gamma, beta,
                                                    out + (size_t)t * NN * CC);
    }
}